// rwkv7MoeSelfAttention_65661460021706
// MI455X (gfx1250) — compile-verified
//
#include <hip/hip_runtime.h>
#include <hip/hip_bf16.h>

#define Bb 2
#define Tt 1024
#define Cc 2048
#define Hh 32
#define Ss 64
#define Aa 2048
#define BT 2048
#define GN_EPS_F 6.4e-4f

typedef __attribute__((ext_vector_type(2))) float v2f;
typedef __attribute__((ext_vector_type(8))) float v8f;
typedef int v4i_vs __attribute__((vector_size(16)));

__device__ __forceinline__ v8f wmma4(v2f a, v2f b, v8f c) {
  // D = A(16x4,f32) x B(4x16,f32) + C(16x16,f32)
  return __builtin_amdgcn_wmma_f32_16x16x4_f32(false, a, false, b, (short)0, c,
                                               false, false);
}

// ---- gfx1250 async global->LDS copy (ASYNCcnt) with sync fallback ----------
#if defined(__has_builtin)
#if __has_builtin(__builtin_amdgcn_global_load_async_to_lds_b128)
#define HAVE_ASYNC_LDS 1
#endif
#endif

__device__ __forceinline__ void cp16(const float* g, float* l) {
#ifdef HAVE_ASYNC_LDS
  __builtin_amdgcn_global_load_async_to_lds_b128(
      (__attribute__((address_space(1))) v4i_vs*)(g),
      (__attribute__((address_space(3))) v4i_vs*)(l), 0, 0);
#else
  *reinterpret_cast<float4*>(l) = *reinterpret_cast<const float4*>(g);
#endif
}

template <int N>
__device__ __forceinline__ void cp_wait() {
#ifdef HAVE_ASYNC_LDS
#if __has_builtin(__builtin_amdgcn_s_wait_asynccnt)
  __builtin_amdgcn_s_wait_asynccnt(N);
#else
  asm volatile("s_wait_asynccnt %0" ::"i"(N) : "memory");
#endif
#endif
}

// ---------------------------------------------------------------------------
// Kernel 1: tmp5 = tanh( (x + (shift-x)*maa_x) @ W1 )   [BT x 160]
// 8 rows per block so W1 is re-used 8x per load (bounds L2 traffic).
// ---------------------------------------------------------------------------
__global__ __launch_bounds__(256) void mix_tmp_kernel(
    const float* __restrict__ hidden, const float* __restrict__ maa_x,
    const float* __restrict__ W1, float* __restrict__ tmp5) {
  int row0 = blockIdx.x * 8;
  __shared__ float xs[8][Cc];  // 64 KB
  for (int i = threadIdx.x; i < 8 * Cc; i += 256) {
    int r = i >> 11, c = i & (Cc - 1);
    int row = row0 + r, t = row % Tt;
    float x = hidden[(size_t)row * Cc + c];
    float sh = (t == 0) ? 0.f : hidden[(size_t)(row - 1) * Cc + c];
    xs[r][c] = x + (sh - x) * maa_x[c];
  }
  __syncthreads();
  if (threadIdx.x < 160) {
    float acc[8] = {0.f, 0.f, 0.f, 0.f, 0.f, 0.f, 0.f, 0.f};
    for (int c = 0; c < Cc; ++c) {
      float w = W1[(size_t)c * 160 + threadIdx.x];
#pragma unroll
      for (int r = 0; r < 8; ++r) acc[r] += xs[r][c] * w;
    }
#pragma unroll
    for (int r = 0; r < 8; ++r)
      tmp5[(size_t)(row0 + r) * 160 + threadIdx.x] = tanhf(acc[r]);
  }
}

// ---------------------------------------------------------------------------
// Kernel 2: build the 5 mixed inputs td_in,k_in,v_in,r_in,g_in (8 rows/block)
// ---------------------------------------------------------------------------
__global__ __launch_bounds__(256) void mix_apply_kernel(
    const float* __restrict__ hidden, const float* __restrict__ maa_w,
    const float* __restrict__ maa_k, const float* __restrict__ maa_v,
    const float* __restrict__ maa_r, const float* __restrict__ maa_g,
    const float* __restrict__ W2, const float* __restrict__ tmp5,
    float* __restrict__ in5) {
  int row0 = blockIdx.x * 8;
  __shared__ float ts[8][160];
  for (int i = threadIdx.x; i < 8 * 160; i += 256) {
    int r = i / 160, n = i % 160;
    ts[r][n] = tmp5[(size_t)(row0 + r) * 160 + n];
  }
  __syncthreads();
  int c = threadIdx.x;
  for (int cg = 0; cg < 8; ++cg, c += 256) {
    float m[8][5];
#pragma unroll
    for (int r = 0; r < 8; ++r)
#pragma unroll
      for (int f = 0; f < 5; ++f) m[r][f] = 0.f;
#pragma unroll
    for (int f = 0; f < 5; ++f) {
      for (int e = 0; e < 32; ++e) {
        float w = W2[((size_t)f * 32 + e) * Cc + c];
#pragma unroll
        for (int r = 0; r < 8; ++r) m[r][f] += ts[r][f * 32 + e] * w;
      }
    }
    float aw = maa_w[c], ak = maa_k[c], av = maa_v[c], ar = maa_r[c],
          ag = maa_g[c];
#pragma unroll
    for (int r = 0; r < 8; ++r) {
      int row = row0 + r, t = row % Tt;
      float x = hidden[(size_t)row * Cc + c];
      float sh = (t == 0) ? 0.f : hidden[(size_t)(row - 1) * Cc + c];
      float xx = sh - x;
      size_t p = (size_t)row * Cc + c;
      in5[p + 0ull * BT * Cc] = x + xx * (aw + m[r][0]);
      in5[p + 1ull * BT * Cc] = x + xx * (ak + m[r][1]);
      in5[p + 2ull * BT * Cc] = x + xx * (av + m[r][2]);
      in5[p + 3ull * BT * Cc] = x + xx * (ar + m[r][3]);
      in5[p + 4ull * BT * Cc] = x + xx * (ag + m[r][4]);
    }
  }
}

// ---------------------------------------------------------------------------
// Kernel 3: fp32 WMMA GEMM   Out[M][N] = X[M][K] @ W[N][K]^T
// 64x64 block tile, 8 waves x two 16x16 accumulators,
// double-buffered async global->LDS K=32 panels.
// ---------------------------------------------------------------------------
__global__ __launch_bounds__(256) void gemm_xwt_kernel(
    const float* __restrict__ X, const float* __restrict__ W,
    float* __restrict__ Out, int M, int N, int K) {
  __shared__ float Xs[2][64][36];
  __shared__ float Ws[2][64][36];
  const int tid = threadIdx.x;
  const int wave = tid >> 5, lane = tid & 31;
  const int half = lane >> 4, lr = lane & 15;
  const int m0 = blockIdx.y * 64, n0 = blockIdx.x * 64;

  const int r0 = tid >> 3;            // 0..31
  const int c40 = (tid & 7) << 2;     // 0,4,..,28

  v8f acc[2];
  acc[0] = (v8f){0.f, 0.f, 0.f, 0.f, 0.f, 0.f, 0.f, 0.f};
  acc[1] = (v8f){0.f, 0.f, 0.f, 0.f, 0.f, 0.f, 0.f, 0.f};

  auto issue = [&](int kc, int bufi) {
    cp16(&X[(size_t)(m0 + r0) * K + kc + c40], &Xs[bufi][r0][c40]);
    cp16(&X[(size_t)(m0 + r0 + 32) * K + kc + c40], &Xs[bufi][r0 + 32][c40]);
    cp16(&W[(size_t)(n0 + r0) * K + kc + c40], &Ws[bufi][r0][c40]);
    cp16(&W[(size_t)(n0 + r0 + 32) * K + kc + c40], &Ws[bufi][r0 + 32][c40]);
  };

  issue(0, 0);
  int buf = 0;
  for (int kc = 0; kc < K; kc += 32) {
    bool more = (kc + 32) < K;
    if (more) issue(kc + 32, buf ^ 1);
    if (more) cp_wait<4>(); else cp_wait<0>();
    __syncthreads();
#pragma unroll
    for (int kk = 0; kk < 32; kk += 4) {
#pragma unroll
      for (int tix = 0; tix < 2; ++tix) {
        int tile = wave * 2 + tix;
        int tr = tile >> 2, tc = tile & 3;
        v2f a, b;
        a.x = Xs[buf][tr * 16 + lr][kk + 2 * half];
        a.y = Xs[buf][tr * 16 + lr][kk + 2 * half + 1];
        b.x = Ws[buf][tc * 16 + lr][kk + 2 * half];
        b.y = Ws[buf][tc * 16 + lr][kk + 2 * half + 1];
        acc[tix] = wmma4(a, b, acc[tix]);
      }
    }
    __syncthreads();
    buf ^= 1;
  }
#pragma unroll
  for (int tix = 0; tix < 2; ++tix) {
    int tile = wave * 2 + tix;
    int tr = tile >> 2, tc = tile & 3;
    int rr0 = m0 + tr * 16 + half * 8;
    int col = n0 + tc * 16 + lr;
#pragma unroll
    for (int i = 0; i < 8; ++i)
      Out[(size_t)(rr0 + i) * N + col] = acc[tix][i];
  }
}

// ---------------------------------------------------------------------------
// Kernel 4: decay  w = exp(-exp(time_decay + tanh(td_in@W1d)@W2d))
// 8 rows per block; activations staged in LDS, weights re-used 8x.
// ---------------------------------------------------------------------------
__global__ __launch_bounds__(256) void decay_kernel(
    const float* __restrict__ td_in, const float* __restrict__ W1d,
    const float* __restrict__ W2d, const float* __restrict__ time_decay,
    float* __restrict__ wdec) {
  int row0 = blockIdx.x * 8;
  __shared__ float td8[8][Cc];      // 64 KB
  __shared__ float part[4][8][64];  // 8 KB
  __shared__ float hs[8][64];
  int tid = threadIdx.x;
  for (int i = tid; i < 8 * Cc; i += 256) {
    int r = i >> 11, c = i & (Cc - 1);
    td8[r][c] = td_in[(size_t)(row0 + r) * Cc + c];
  }
  __syncthreads();
  int e = tid & 63, q = tid >> 6;
  {
    float acc[8] = {0.f, 0.f, 0.f, 0.f, 0.f, 0.f, 0.f, 0.f};
    for (int c = q * 512; c < q * 512 + 512; ++c) {
      float w = W1d[(size_t)c * 64 + e];
#pragma unroll
      for (int r = 0; r < 8; ++r) acc[r] += td8[r][c] * w;
    }
#pragma unroll
    for (int r = 0; r < 8; ++r) part[q][r][e] = acc[r];
  }
  __syncthreads();
  if (tid < 64) {
#pragma unroll
    for (int r = 0; r < 8; ++r)
      hs[r][tid] = tanhf(part[0][r][tid] + part[1][r][tid] + part[2][r][tid] +
                         part[3][r][tid]);
  }
  __syncthreads();
  for (int cg = 0; cg < 8; ++cg) {
    int a = cg * 256 + tid;
    float td0 = time_decay[a];
    float acc2[8];
#pragma unroll
    for (int r = 0; r < 8; ++r) acc2[r] = td0;
    for (int ee = 0; ee < 64; ++ee) {
      float w = W2d[(size_t)ee * Aa + a];
#pragma unroll
      for (int r = 0; r < 8; ++r) acc2[r] += hs[r][ee] * w;
    }
#pragma unroll
    for (int r = 0; r < 8; ++r)
      wdec[(size_t)(row0 + r) * Aa + a] = expf(-expf(acc2[r]));
  }
}

// ---------------------------------------------------------------------------
// Kernel 5: chunked RWKV scan with fp32 WMMA.  One block per (b,h).
// Chunk L=16:  O = (r*Wt)@S0 + causal[(r*Wt)@(k/Ws1)^T]@V ;
//              S0 <- diag(WL)S0 + (WL*k/Ws1)^T @ V
// Next chunk's r/k/v/w stream in via async global->LDS while computing.
// ---------------------------------------------------------------------------
#define SP 68
__global__ __launch_bounds__(256) void scan_kernel(
    const float* __restrict__ rbuf, const float* __restrict__ kbuf,
    const float* __restrict__ vbuf, const float* __restrict__ wdec,
    const float* __restrict__ time_faaaa, const float* __restrict__ init_state,
    float* __restrict__ ybuf) {
  int bh = blockIdx.x;
  int b = bh >> 5, h = bh & 31;
  const int tid = threadIdx.x;
  const int wave = tid >> 5, lane = tid & 31;
  const int half = lane >> 4, lr = lane & 15;

  __shared__ float S0[64][SP];
  __shared__ float rr[2][16][SP], kr[2][16][SP], vv[2][16][SP], ww[2][16][SP];
  __shared__ float rt[16][SP], kt[16][SP], kL[16][SP];
  __shared__ float Wc[17][64];
  __shared__ float Amat[16][20], Amask[16][20];
  __shared__ float ddiag[16];
  __shared__ float Om[16][SP];
  __shared__ float uu[64];

  auto issueChunk = [&](int ci2, int bufi) {
    size_t rowbase2 = (size_t)b * Tt + ci2 * 16;
    int t2 = tid >> 4, c4 = (tid & 15) << 2;
    size_t g = (rowbase2 + t2) * Aa + (size_t)h * 64 + c4;
    cp16(&rbuf[g], &rr[bufi][t2][c4]);
    cp16(&kbuf[g], &kr[bufi][t2][c4]);
    cp16(&vbuf[g], &vv[bufi][t2][c4]);
    cp16(&wdec[g], &ww[bufi][t2][c4]);
  };

  issueChunk(0, 0);
  if (tid < 64) uu[tid] = time_faaaa[h * 64 + tid];
  for (int i = tid; i < 4096; i += 256) {
    int kkk = i >> 6, vvv = i & 63;
    S0[kkk][vvv] = init_state[(((size_t)b * Hh + h) * Ss + kkk) * Ss + vvv];
  }

  int buf = 0;
  for (int ci = 0; ci < Tt / 16; ++ci) {
    size_t rowbase = (size_t)b * Tt + ci * 16;
    bool more = (ci + 1) < (Tt / 16);
    if (more) issueChunk(ci + 1, buf ^ 1);
    if (more) cp_wait<4>(); else cp_wait<0>();
    __syncthreads();
    // ---- per-channel cumulative decay ----
    if (tid < 64) {
      float p = 1.f;
#pragma unroll
      for (int t = 0; t < 16; ++t) { Wc[t][tid] = p; p *= ww[buf][t][tid]; }
      Wc[16][tid] = p;
    }
    __syncthreads();
    // ---- scaled operands ----
    for (int i = tid; i < 1024; i += 256) {
      int t = i >> 6, c = i & 63;
      float wn = Wc[t][c], wd = Wc[t + 1][c], wl = Wc[16][c];
      rt[t][c] = rr[buf][t][c] * wn;
      float ktv = kr[buf][t][c] / wd;
      kt[t][c] = ktv;
      kL[t][c] = ktv * wl;
    }
    __syncthreads();
    // ---- O_inter (waves 0-3), A (wave 4), diag (wave 5) ----
    if (wave < 4) {
      int vc = wave;
      v8f c = (v8f){0.f, 0.f, 0.f, 0.f, 0.f, 0.f, 0.f, 0.f};
#pragma unroll
      for (int kk = 0; kk < 64; kk += 4) {
        v2f a, bb;
        a.x = rt[lr][kk + 2 * half];
        a.y = rt[lr][kk + 2 * half + 1];
        bb.x = S0[kk + 2 * half][vc * 16 + lr];
        bb.y = S0[kk + 2 * half + 1][vc * 16 + lr];
        c = wmma4(a, bb, c);
      }
#pragma unroll
      for (int i = 0; i < 8; ++i) Om[half * 8 + i][vc * 16 + lr] = c[i];
    } else if (wave == 4) {
      v8f c = (v8f){0.f, 0.f, 0.f, 0.f, 0.f, 0.f, 0.f, 0.f};
#pragma unroll
      for (int kk = 0; kk < 64; kk += 4) {
        v2f a, bb;
        a.x = rt[lr][kk + 2 * half];
        a.y = rt[lr][kk + 2 * half + 1];
        bb.x = kt[lr][kk + 2 * half];  // B[k][s] = kt[s][k]
        bb.y = kt[lr][kk + 2 * half + 1];
        c = wmma4(a, bb, c);
      }
#pragma unroll
      for (int i = 0; i < 8; ++i) Amat[half * 8 + i][lr] = c[i];
    } else if (wave == 5) {
      if (lane < 16) {
        float acc = 0.f;
        for (int k2 = 0; k2 < 64; ++k2)
          acc += rr[buf][lane][k2] * uu[k2] * kr[buf][lane][k2];
        ddiag[lane] = acc;
      }
    }
    __syncthreads();
    // ---- causal mask + diagonal ----
    {
      int t = tid >> 4, s = tid & 15;
      float av = Amat[t][s];
      Amask[t][s] = (s < t) ? av : ((s == t) ? ddiag[t] : 0.f);
    }
    __syncthreads();
    // ---- O_intra + emit (waves 0-3); state update (all waves, 2 tiles) ----
    if (wave < 4) {
      int vc = wave;
      v8f c;
#pragma unroll
      for (int i = 0; i < 8; ++i) c[i] = Om[half * 8 + i][vc * 16 + lr];
#pragma unroll
      for (int kk = 0; kk < 16; kk += 4) {
        v2f a, bb;
        a.x = Amask[lr][kk + 2 * half];
        a.y = Amask[lr][kk + 2 * half + 1];
        bb.x = vv[buf][kk + 2 * half][vc * 16 + lr];
        bb.y = vv[buf][kk + 2 * half + 1][vc * 16 + lr];
        c = wmma4(a, bb, c);
      }
#pragma unroll
      for (int i = 0; i < 8; ++i) {
        int t = half * 8 + i;
        ybuf[(rowbase + t) * Aa + (size_t)h * 64 + vc * 16 + lr] = c[i];
      }
    }
#pragma unroll
    for (int tji = 0; tji < 2; ++tji) {
      int tile = wave * 2 + tji;
      int krq = tile >> 2, vc = tile & 3;
      v8f c;
#pragma unroll
      for (int i = 0; i < 8; ++i) {
        int krow = krq * 16 + half * 8 + i;
        c[i] = Wc[16][krow] * S0[krow][vc * 16 + lr];
      }
#pragma unroll
      for (int kk = 0; kk < 16; kk += 4) {
        v2f a, bb;
        a.x = kL[kk + 2 * half][krq * 16 + lr];  // A = kL^T
        a.y = kL[kk + 2 * half + 1][krq * 16 + lr];
        bb.x = vv[buf][kk + 2 * half][vc * 16 + lr];
        bb.y = vv[buf][kk + 2 * half + 1][vc * 16 + lr];
        c = wmma4(a, bb, c);
      }
#pragma unroll
      for (int i = 0; i < 8; ++i)
        S0[krq * 16 + half * 8 + i][vc * 16 + lr] = c[i];
    }
    __syncthreads();
    buf ^= 1;
  }
}

// ---------------------------------------------------------------------------
// Kernel 6: groupnorm(64) * ln_w + ln_b, then * silu(g)
// ---------------------------------------------------------------------------
__global__ __launch_bounds__(256) void gn_gate_kernel(
    const float* __restrict__ y, const float* __restrict__ graw,
    const float* __restrict__ ln_w, const float* __restrict__ ln_b,
    float* __restrict__ y2) {
  int row = blockIdx.x;
  int tid = threadIdx.x;
  __shared__ float ys[Cc];
  __shared__ float ps[256], pq[256];
  __shared__ float mu[32], rs[32];
  for (int c = tid; c < Cc; c += 256) ys[c] = y[(size_t)row * Cc + c];
  __syncthreads();
  int hh = tid >> 3, sub = tid & 7;
  float s = 0.f, sq = 0.f;
#pragma unroll
  for (int i = 0; i < 8; ++i) {
    float v = ys[hh * 64 + sub * 8 + i];
    s += v;
    sq += v * v;
  }
  ps[tid] = s;
  pq[tid] = sq;
  __syncthreads();
  if (tid < 32) {
    float S = 0.f, Q = 0.f;
#pragma unroll
    for (int j = 0; j < 8; ++j) { S += ps[tid * 8 + j]; Q += pq[tid * 8 + j]; }
    float m = S * (1.f / 64.f);
    mu[tid] = m;
    rs[tid] = rsqrtf(Q * (1.f / 64.f) - m * m + GN_EPS_F);
  }
  __syncthreads();
  for (int c = tid; c < Cc; c += 256) {
    int h2 = c >> 6;
    float val = (ys[c] - mu[h2]) * rs[h2] * ln_w[c] + ln_b[c];
    float g = graw[(size_t)row * Cc + c];
    val *= g / (1.f + expf(-g));
    y2[(size_t)row * Cc + c] = val;
  }
}

// ---------------------------------------------------------------------------
extern "C" void kernel_launch(void* const* d_in, const int* in_sizes, int n_in,
                              void* d_out, int out_size, void* d_ws,
                              size_t ws_size, hipStream_t stream) {
  const float* hidden     = (const float*)d_in[0];
  const float* init_state = (const float*)d_in[1];
  const float* maa_x      = (const float*)d_in[2];
  const float* maa_w      = (const float*)d_in[3];
  const float* maa_k      = (const float*)d_in[4];
  const float* maa_v      = (const float*)d_in[5];
  const float* maa_r      = (const float*)d_in[6];
  const float* maa_g      = (const float*)d_in[7];
  const float* maa_w1     = (const float*)d_in[8];
  const float* maa_w2     = (const float*)d_in[9];
  const float* time_decay = (const float*)d_in[10];
  const float* dec_w1     = (const float*)d_in[11];
  const float* dec_w2     = (const float*)d_in[12];
  const float* time_faaaa = (const float*)d_in[13];
  const float* Wr         = (const float*)d_in[14];
  const float* Wk         = (const float*)d_in[15];
  const float* Wv         = (const float*)d_in[16];
  const float* Wg         = (const float*)d_in[17];
  const float* Wo         = (const float*)d_in[18];
  const float* ln_w       = (const float*)d_in[19];
  const float* ln_b       = (const float*)d_in[20];

  float* ws   = (float*)d_ws;
  float* tmp5 = ws;                                   // BT*160
  float* in5  = tmp5 + (size_t)BT * 160;              // 5*BT*C
  float* rO   = in5 + 5ull * BT * Cc;                 // BT*A each below
  float* kO   = rO + (size_t)BT * Aa;
  float* vO   = kO + (size_t)BT * Aa;
  float* gO   = vO + (size_t)BT * Aa;
  float* wdec = gO + (size_t)BT * Aa;
  float* ybuf = wdec + (size_t)BT * Aa;
  float* y2   = ybuf + (size_t)BT * Aa;

  const float* td_in = in5 + 0ull * BT * Cc;
  const float* k_in  = in5 + 1ull * BT * Cc;
  const float* v_in  = in5 + 2ull * BT * Cc;
  const float* r_in  = in5 + 3ull * BT * Cc;
  const float* g_in  = in5 + 4ull * BT * Cc;

  mix_tmp_kernel<<<BT / 8, 256, 0, stream>>>(hidden, maa_x, maa_w1, tmp5);
  mix_apply_kernel<<<BT / 8, 256, 0, stream>>>(hidden, maa_w, maa_k, maa_v,
                                               maa_r, maa_g, maa_w2, tmp5, in5);

  dim3 gg(Aa / 64, BT / 64);
  gemm_xwt_kernel<<<gg, 256, 0, stream>>>(r_in, Wr, rO, BT, Aa, Cc);
  gemm_xwt_kernel<<<gg, 256, 0, stream>>>(k_in, Wk, kO, BT, Aa, Cc);
  gemm_xwt_kernel<<<gg, 256, 0, stream>>>(v_in, Wv, vO, BT, Aa, Cc);
  gemm_xwt_kernel<<<gg, 256, 0, stream>>>(g_in, Wg, gO, BT, Aa, Cc);

  decay_kernel<<<BT / 8, 256, 0, stream>>>(td_in, dec_w1, dec_w2, time_decay,
                                           wdec);

  scan_kernel<<<Bb * Hh, 256, 0, stream>>>(rO, kO, vO, wdec, time_faaaa,
                                           init_state, ybuf);

  gn_gate_kernel<<<BT, 256, 0, stream>>>(ybuf, gO, ln_w, ln_b, y2);

  dim3 go(Cc / 64, BT / 64);
  gemm_xwt_kernel<<<go, 256, 0, stream>>>(y2, Wo, (float*)d_out, BT, Cc, Aa);
}